// TRLN_7945689497742
// MI455X (gfx1250) — compile-verified
//
#include <hip/hip_runtime.h>
#include <math.h>

typedef float v2f __attribute__((ext_vector_type(2)));
typedef float v8f __attribute__((ext_vector_type(8)));

#define BN_EPS 1e-3f
#define PD_EPS 1e-6f

#define BATCH 1024
#define CH    60
#define SW    288
#define F1    8
#define FD    16      // F1*D
#define F2    16
#define KL    64
#define NCLS  4
#define W4    72
#define W9    9
#define FEATD 144
#define UPITCH 352    // 288 + 2*32 halo
#define PPITCH 80     // 72 padded to 5*16 for WMMA

// workspace layout (float offsets)
#define OFF_CONST 0
#define OFF_FEAT  64
#define OFF_CENT  (64 + BATCH*FEATD)
#define OFF_DIST  (OFF_CENT + NCLS*FEATD)

// ---------------------------------------------------------------- precompute
// Fold BN1+BN2 (+ depthwise row-sum for the BN1 bias term) and BN3 into
// per-channel scale/bias: cst[0..15]=A2, [16..31]=B2, [32..47]=s3, [48..63]=t3.
__global__ void eegnet_pre(const float* __restrict__ dw,
                           const float* __restrict__ g1, const float* __restrict__ b1,
                           const float* __restrict__ m1, const float* __restrict__ v1,
                           const float* __restrict__ g2, const float* __restrict__ b2,
                           const float* __restrict__ m2, const float* __restrict__ v2,
                           const float* __restrict__ g3, const float* __restrict__ b3,
                           const float* __restrict__ m3, const float* __restrict__ v3,
                           float* __restrict__ cst) {
  int j = threadIdx.x;
  if (j < FD) {
    int f = j >> 1;
    float s1 = g1[f] * rsqrtf(v1[f] + BN_EPS);
    float t1 = b1[f] - m1[f] * s1;
    float s2 = g2[j] * rsqrtf(v2[j] + BN_EPS);
    float t2 = b2[j] - m2[j] * s2;
    float sum = 0.f;
    for (int c = 0; c < CH; ++c) sum += dw[j * CH + c];
    cst[j]      = s2 * s1;                 // scale on width-conv output
    cst[16 + j] = s2 * t1 * sum + t2;      // fused bias
    float s3 = g3[j] * rsqrtf(v3[j] + BN_EPS);
    cst[32 + j] = s3;
    cst[48 + j] = b3[j] - m3[j] * s3;
  }
}

// ---------------------------------------------------------------- main fused
__global__ __launch_bounds__(128) void eegnet_main(
    const float* __restrict__ x, const float* __restrict__ conv1_w,
    const float* __restrict__ dw_w, const float* __restrict__ pw_w,
    const float* __restrict__ cst, float* __restrict__ feat) {
  __shared__ float dwS[FD * CH];       // 960
  __shared__ float cwS[F1 * KL];       // 512
  __shared__ float pwS[F2 * FD];       // 256
  __shared__ float cS[64];
  __shared__ float uS[FD * UPITCH];    // 5632 (u with +-32 zero halo)
  __shared__ float pS[FD * PPITCH];    // 1280 (post pool4, zero-padded cols)
  __shared__ float qS[F2 * W4];        // 1152 (post BN3+ELU)

  const int tid = threadIdx.x;
  const int b   = blockIdx.x;

  for (int i = tid; i < FD * CH; i += 128) dwS[i] = dw_w[i];
  for (int i = tid; i < F1 * KL; i += 128) cwS[i] = conv1_w[i];
  for (int i = tid; i < F2 * FD; i += 128) pwS[i] = pw_w[i];
  if (tid < 64) cS[tid] = cst[tid];
  for (int i = tid; i < FD * 64; i += 128) {            // zero u halo
    int j = i >> 6, k = i & 63;
    uS[j * UPITCH + ((k < 32) ? k : (k + 288))] = 0.f;
  }
  __syncthreads();

  const int lane = tid & 31;
  const int wave = tid >> 5;
  const int m    = lane & 15;   // M row (A/C) or N col (B/C)
  const int hi   = lane >> 4;   // half-wave select

  // ---- step 1: u[16,288] = dw[16,60] @ x_b[60,288]  (WMMA f32 16x16x4)
  const float* xb = x + (size_t)b * (CH * SW);
  for (int t = wave; t < 18; t += 4) {
    int n0 = t * 16;
    v8f acc = {};
    for (int kc = 0; kc < CH; kc += 4) {
      int ka = kc + 2 * hi;
      v2f a, bb;
      a.x  = dwS[m * CH + ka];
      a.y  = dwS[m * CH + ka + 1];
      bb.x = xb[ka * SW + n0 + m];
      bb.y = xb[(ka + 1) * SW + n0 + m];
      acc = __builtin_amdgcn_wmma_f32_16x16x4_f32(false, a, false, bb,
                                                  (short)0, acc, false, false);
    }
#pragma unroll
    for (int r = 0; r < 8; ++r)
      uS[(r + 8 * hi) * UPITCH + 32 + n0 + m] = acc[r];
  }
  __syncthreads();

  // ---- step 2: width conv (K=64, pad 32) + BN2 + ELU + avgpool4 -> pS[16,72]
  for (int i = tid; i < FD * 8; i += 128)               // zero pad cols 72..79
    pS[(i >> 3) * PPITCH + 72 + (i & 7)] = 0.f;
  for (int idx = tid; idx < FD * W4; idx += 128) {
    int j = idx / W4, w4 = idx % W4;
    const float* cw = &cwS[(j >> 1) * KL];
    const float* uj = &uS[j * UPITCH];
    float A2 = cS[j], B2 = cS[16 + j];
    float acc = 0.f;
#pragma unroll
    for (int d = 0; d < 4; ++d) {
      int w = 4 * w4 + d;                 // conv out index (w=288 unused by pool)
      float s = 0.f;
#pragma unroll
      for (int k = 0; k < KL; ++k) s = fmaf(cw[k], uj[w + k], s);
      float z = fmaf(A2, s, B2);
      acc += (z > 0.f) ? z : expm1f(z);   // ELU(alpha=1)
    }
    pS[j * PPITCH + w4] = acc * 0.25f;
  }
  __syncthreads();

  // ---- step 3: pointwise pw[16,16] @ p[16,80] (WMMA) + BN3 + ELU -> qS[16,72]
  for (int t = wave; t < 5; t += 4) {
    int n0 = t * 16;
    v8f acc = {};
    for (int kc = 0; kc < FD; kc += 4) {
      int ka = kc + 2 * hi;
      v2f a, bb;
      a.x  = pwS[m * FD + ka];
      a.y  = pwS[m * FD + ka + 1];
      bb.x = pS[ka * PPITCH + n0 + m];
      bb.y = pS[(ka + 1) * PPITCH + n0 + m];
      acc = __builtin_amdgcn_wmma_f32_16x16x4_f32(false, a, false, bb,
                                                  (short)0, acc, false, false);
    }
    int w4 = n0 + m;
    if (w4 < W4) {
#pragma unroll
      for (int r = 0; r < 8; ++r) {
        int o = r + 8 * hi;
        float z = fmaf(cS[32 + o], acc[r], cS[48 + o]);
        qS[o * W4 + w4] = (z > 0.f) ? z : expm1f(z);
      }
    }
  }
  __syncthreads();

  // ---- step 4: avgpool8 -> feat[b,144]
  for (int i = tid; i < FEATD; i += 128) {
    int o = i / W9, w9 = i % W9;
    float s = 0.f;
#pragma unroll
    for (int k = 0; k < 8; ++k) s += qS[o * W4 + 8 * w9 + k];
    feat[(size_t)b * FEATD + i] = s * 0.125f;
  }
}

// ---------------------------------------------------------------- centroids
__global__ void eegnet_cent(const float* __restrict__ feat,
                            const int* __restrict__ y, float* __restrict__ cent) {
  int tid = threadIdx.x;                 // 576 = 4 classes * 144 dims
  int cls = tid / FEATD, d = tid % FEATD;
  float s = 0.f, cnt = 0.f;
  for (int b = 0; b < BATCH; ++b)
    if (y[b] == cls) { s += feat[b * FEATD + d]; cnt += 1.f; }
  cent[cls * FEATD + d] = s / fmaxf(cnt, 1.f);
}

// ---------------------------------------------------------------- dist + softmax
__global__ void eegnet_head(const float* __restrict__ feat,
                            const int* __restrict__ y,
                            const float* __restrict__ cent,
                            const float* __restrict__ fcw,
                            const float* __restrict__ fcb,
                            float* __restrict__ probs, float* __restrict__ dist) {
  int b = blockIdx.x * blockDim.x + threadIdx.x;
  if (b >= BATCH) return;
  const float* f = feat + (size_t)b * FEATD;
  const float* c = cent + y[b] * FEATD;
  float ss = 0.f;
  float lg[NCLS] = {0.f, 0.f, 0.f, 0.f};
  for (int d = 0; d < FEATD; ++d) {
    float fv = f[d];
    float df = fv - c[d] + PD_EPS;
    ss = fmaf(df, df, ss);
#pragma unroll
    for (int k = 0; k < NCLS; ++k) lg[k] = fmaf(fv, fcw[k * FEATD + d], lg[k]);
  }
  dist[b] = sqrtf(ss);
  float mx = -1e30f;
#pragma unroll
  for (int k = 0; k < NCLS; ++k) { lg[k] += fcb[k]; mx = fmaxf(mx, lg[k]); }
  float se = 0.f;
#pragma unroll
  for (int k = 0; k < NCLS; ++k) { lg[k] = expf(lg[k] - mx); se += lg[k]; }
  float inv = 1.f / se;
#pragma unroll
  for (int k = 0; k < NCLS; ++k) probs[b * NCLS + k] = lg[k] * inv;
}

// ---------------------------------------------------------------- loss mean
__global__ void eegnet_mean(const float* __restrict__ dist, float* __restrict__ out) {
  __shared__ float red[256];
  int tid = threadIdx.x;
  float s = 0.f;
  for (int i = tid; i < BATCH; i += 256) s += dist[i];
  red[tid] = s;
  __syncthreads();
  for (int st = 128; st > 0; st >>= 1) {
    if (tid < st) red[tid] += red[tid + st];
    __syncthreads();
  }
  if (tid == 0) out[0] = red[0] * (1.f / BATCH);
}

extern "C" void kernel_launch(void* const* d_in, const int* in_sizes, int n_in,
                              void* d_out, int out_size, void* d_ws, size_t ws_size,
                              hipStream_t stream) {
  const float* x       = (const float*)d_in[0];
  const int*   y       = (const int*)  d_in[1];
  const float* conv1_w = (const float*)d_in[2];
  const float* bn1g = (const float*)d_in[3];
  const float* bn1b = (const float*)d_in[4];
  const float* bn1m = (const float*)d_in[5];
  const float* bn1v = (const float*)d_in[6];
  const float* dw_w = (const float*)d_in[7];
  const float* bn2g = (const float*)d_in[8];
  const float* bn2b = (const float*)d_in[9];
  const float* bn2m = (const float*)d_in[10];
  const float* bn2v = (const float*)d_in[11];
  const float* pw_w = (const float*)d_in[12];
  const float* bn3g = (const float*)d_in[13];
  const float* bn3b = (const float*)d_in[14];
  const float* bn3m = (const float*)d_in[15];
  const float* bn3v = (const float*)d_in[16];
  const float* fcw  = (const float*)d_in[17];
  const float* fcb  = (const float*)d_in[18];

  float* ws   = (float*)d_ws;
  float* cst  = ws + OFF_CONST;
  float* feat = ws + OFF_FEAT;
  float* cent = ws + OFF_CENT;
  float* dist = ws + OFF_DIST;
  float* out  = (float*)d_out;

  eegnet_pre<<<1, 32, 0, stream>>>(dw_w, bn1g, bn1b, bn1m, bn1v,
                                   bn2g, bn2b, bn2m, bn2v,
                                   bn3g, bn3b, bn3m, bn3v, cst);
  eegnet_main<<<BATCH, 128, 0, stream>>>(x, conv1_w, dw_w, pw_w, cst, feat);
  eegnet_cent<<<1, NCLS * FEATD, 0, stream>>>(feat, y, cent);
  eegnet_head<<<8, 128, 0, stream>>>(feat, y, cent, fcw, fcb, out, dist);
  eegnet_mean<<<1, 256, 0, stream>>>(dist, out + BATCH * NCLS);
}